// DendriticFullyConnected_21603685499097
// MI455X (gfx1250) — compile-verified
//
#include <hip/hip_runtime.h>

// CDNA5 (gfx1250) wave32 WMMA fragments
typedef __attribute__((ext_vector_type(2))) float v2f;  // A/B frag for 16x16x4 f32
typedef __attribute__((ext_vector_type(8))) float v8f;  // C/D frag (16x16 f32)

#define B_DIM   128
#define IN_DIM  1024
#define OUT_DIM 1024

// One wave computes one 16x16 output tile of BOTH fused GEMMs:
//   accN = x @ W_non^T      (for state = sigmoid(accN + b) - 1)
//   accM = x' @ W_nmda^T    (x' = x with k=0 / k=IN-1 scaled 0.5, folded into B)
// then applies the Hill epilogue and writes y.
__global__ __launch_bounds__(256) void dendritic_fused_kernel(
    const float* __restrict__ X,      // [B, IN]
    const float* __restrict__ Wnmda,  // [OUT, IN]
    const float* __restrict__ Wnon,   // [OUT, IN]
    const float* __restrict__ bnon,   // [OUT]
    float* __restrict__ Y)            // [B, OUT]
{
    const int lane = threadIdx.x & 31;
    const int wave = threadIdx.x >> 5;
    const int tile = blockIdx.x * 8 + wave;   // 64 blocks * 8 waves = 512 tiles
    const int mt = tile >> 6;                 // 0..7   (128/16 row tiles)
    const int nt = tile & 63;                 // 0..63  (1024/16 col tiles)

    const int mrow = mt * 16 + (lane & 15);   // A-matrix row held by this lane
    const int nrow = nt * 16 + (lane & 15);   // B-matrix (W row) held by this lane
    const int h4   = (lane >> 4) * 4;         // lane-half K sub-offset (permuted K)

    const float* xr = X     + (size_t)mrow * IN_DIM + h4;
    const float* wn = Wnon  + (size_t)nrow * IN_DIM + h4;
    const float* wm = Wnmda + (size_t)nrow * IN_DIM + h4;

    v8f accN = {0.f,0.f,0.f,0.f,0.f,0.f,0.f,0.f};
    v8f accM = {0.f,0.f,0.f,0.f,0.f,0.f,0.f,0.f};

    // conv [0.5,0.5] boundary weights: k==0 lives in lanes<16 (.x of kb==0 chunk),
    // k==IN-1 lives in lanes>=16 (.w of the last chunk).
    const float edge_lo = (lane < 16)  ? 0.5f : 1.0f;
    const float edge_hi = (lane >= 16) ? 0.5f : 1.0f;

    for (int kb = 0; kb < IN_DIM; kb += 8) {
        float4 a4  = *(const float4*)(xr + kb);   // global_load_b128
        float4 b4n = *(const float4*)(wn + kb);
        float4 b4m = *(const float4*)(wm + kb);
        if (kb == 0)            b4m.x *= edge_lo;
        if (kb == IN_DIM - 8)   b4m.w *= edge_hi;

        v2f a0  = {a4.x,  a4.y},  a1  = {a4.z,  a4.w};
        v2f bn0 = {b4n.x, b4n.y}, bn1 = {b4n.z, b4n.w};
        v2f bm0 = {b4m.x, b4m.y}, bm1 = {b4m.z, b4m.w};

        // 8 args: (neg_a, A, neg_b, B, c_mod, C, reuse_a, reuse_b)
        accN = __builtin_amdgcn_wmma_f32_16x16x4_f32(false, a0, false, bn0, (short)0, accN, false, false);
        accN = __builtin_amdgcn_wmma_f32_16x16x4_f32(false, a1, false, bn1, (short)0, accN, false, false);
        accM = __builtin_amdgcn_wmma_f32_16x16x4_f32(false, a0, false, bm0, (short)0, accM, false, false);
        accM = __builtin_amdgcn_wmma_f32_16x16x4_f32(false, a1, false, bm1, (short)0, accM, false, false);
    }

    // C/D layout: VGPR i -> row i (lanes 0-15) / row i+8 (lanes 16-31), col = lane&15
    const float bb = bnon[nrow];
    float* op = Y + (size_t)(mt * 16 + (lane >> 4) * 8) * OUT_DIM + nrow;
    #pragma unroll
    for (int i = 0; i < 8; ++i) {
        float z     = accN[i] + bb;
        float s     = 1.0f / (1.0f + __expf(-z));   // sigmoid
        float state = s - 1.0f;                     // in (-1, 0)
        float ca    = fmaxf(accM[i], 0.0f);         // relu(cluster activity)
        float nm    = 2.0f + state;                 // Hill n modulated
        float kd    = exp2f(-nm);                   // 0.5^nm
        float xn    = __powf(ca, nm);               // ca^nm (ca==0 -> 0)
        float y     = xn / (kd + xn) + state;
        op[(size_t)i * OUT_DIM] = y;
    }
}

extern "C" void kernel_launch(void* const* d_in, const int* in_sizes, int n_in,
                              void* d_out, int out_size, void* d_ws, size_t ws_size,
                              hipStream_t stream) {
    const float* X     = (const float*)d_in[0];  // inputs [128,1024]
    const float* Wnmda = (const float*)d_in[1];  // [1024,1024]
    const float* Wnon  = (const float*)d_in[2];  // [1024,1024]
    const float* bnon  = (const float*)d_in[3];  // [1024]
    float* Y = (float*)d_out;                    // [128,1024]

    dim3 grid(64);    // 512 tiles / 8 waves per block
    dim3 block(256);  // 8 wave32 waves
    hipLaunchKernelGGL(dendritic_fused_kernel, grid, block, 0, stream,
                       X, Wnmda, Wnon, bnon, Y);
}